// HGCN_20186346291925
// MI455X (gfx1250) — compile-verified
//
#include <hip/hip_runtime.h>
#include <math.h>

// ---------------------------------------------------------------------------
// HGCN decoder on MI455X (gfx1250)
//  - fp32 WMMA 16x16x4 GEMMs
//  - async global->LDS staging (GLOBAL_LOAD_ASYNC_TO_LDS_*, ASYNCcnt) with
//    double-buffered K-blocks
//  - logmap0 scale folded into each producer: buffers hold tangent vectors,
//    so all GEMM stagings are pure copies
// dims: B=512, N=128, D=256, F=16, L=3
// ---------------------------------------------------------------------------

typedef __attribute__((ext_vector_type(2))) float v2f;
typedef __attribute__((ext_vector_type(8))) float v8f;

#define NB    512
#define NN    128
#define ND    256
#define NF    16
#define NROWS (NB * NN)          // 65536
#define EPSC  1e-7f
#define MAXN  (1.0f - 1e-5f)

__device__ __forceinline__ v8f wmma_f32(v2f a, v2f b, v8f c) {
    return __builtin_amdgcn_wmma_f32_16x16x4_f32(
        false, a, false, b, (short)0, c, false, false);
}

__device__ __forceinline__ float atanh_fast(float x) {   // x in [0, MAXN]
    return 0.5f * logf((1.0f + x) / (1.0f - x));
}

// --- CDNA5 async global->LDS copies (tracked on ASYNCcnt) -------------------
__device__ __forceinline__ void async_b128(unsigned int lds_off, const float* g) {
    asm volatile("global_load_async_to_lds_b128 %0, %1, off"
                 :: "v"(lds_off), "v"(g) : "memory");
}
__device__ __forceinline__ void async_b32(unsigned int lds_off, const float* g) {
    asm volatile("global_load_async_to_lds_b32 %0, %1, off"
                 :: "v"(lds_off), "v"(g) : "memory");
}
#define LDS_OFF(p) ((unsigned int)(uintptr_t)(p))
#define WAIT_ASYNC(n) asm volatile("s_wait_asynccnt " #n ::: "memory")

// ---------------------------------------------------------------------------
// Kernel 0: tan = logmap0(proj(x))   (one wave32 per row)
// ---------------------------------------------------------------------------
__global__ void __launch_bounds__(256) init_tangent(
    const float* __restrict__ x, float* __restrict__ tanv)
{
    int row  = blockIdx.x * 8 + (threadIdx.x >> 5);
    int lane = threadIdx.x & 31;
    const float* xr = x + (size_t)row * ND;
    float vals[8];
    float ss = 0.f;
#pragma unroll
    for (int i = 0; i < 8; ++i) {
        vals[i] = xr[lane + i * 32];
        ss += vals[i] * vals[i];
    }
#pragma unroll
    for (int off = 1; off < 32; off <<= 1)
        ss += __shfl_xor(ss, off, 32);

    float nx = sqrtf(ss);                          // ||x||
    float n1 = fmaxf(nx, EPSC);
    float sp = (n1 > MAXN) ? (MAXN / n1) : 1.0f;   // proj scale
    float nh = nx * sp;                            // ||proj(x)||
    float n2 = fmaxf(nh, EPSC);
    float ls = atanh_fast(fminf(n2, MAXN)) / n2;   // logmap0 scale
    float sc = sp * ls;

    float* tr = tanv + (size_t)row * ND;
#pragma unroll
    for (int i = 0; i < 8; ++i)
        tr[lane + i * 32] = vals[i] * sc;
}

// ---------------------------------------------------------------------------
// Kernel 1: t1 = tan @ W + bias      (pure GEMM, async double-buffered)
// block 256 thr / 8 waves; tile 64 x 128; K-block 32; wave = 2x2 WMMA tiles
// async per K-block: 2x A b128 + 4x B b128 = 6 issues/wave
// ---------------------------------------------------------------------------
__global__ void __launch_bounds__(256) tangent_linear(
    const float* __restrict__ tanv, const float* __restrict__ W,
    const float* __restrict__ bias, float* __restrict__ t1)
{
    const int row0 = blockIdx.x * 64;
    const int col0 = blockIdx.y * 128;
    const int tid  = threadIdx.x;
    const int wave = tid >> 5;
    const int lane = tid & 31;
    const int wrow = (wave & 1) * 32;
    const int wcol = (wave >> 1) * 32;

    __shared__ float Als[2][64][36];     // 64x32, stride 36 (16B-aligned rows)
    __shared__ float Bls[2][32][144];    // 32x128, stride 144

    // per-thread copy chunks (4 floats each)
    int ar[2], ac[2], br[4], bc[4];
#pragma unroll
    for (int i = 0; i < 2; ++i) { int ch = tid + i * 256; ar[i] = ch >> 3; ac[i] = (ch & 7) * 4; }
#pragma unroll
    for (int i = 0; i < 4; ++i) { int ch = tid + i * 256; br[i] = ch >> 5; bc[i] = (ch & 31) * 4; }

#define PREFETCH_K1(buf, k0)                                                   \
    {                                                                          \
        _Pragma("unroll")                                                      \
        for (int i = 0; i < 2; ++i)                                            \
            async_b128(LDS_OFF(&Als[buf][ar[i]][ac[i]]),                       \
                       tanv + (size_t)(row0 + ar[i]) * ND + (k0) + ac[i]);     \
        _Pragma("unroll")                                                      \
        for (int i = 0; i < 4; ++i)                                            \
            async_b128(LDS_OFF(&Bls[buf][br[i]][bc[i]]),                       \
                       W + (size_t)((k0) + br[i]) * ND + col0 + bc[i]);        \
    }

    v8f acc00 = {}, acc01 = {}, acc10 = {}, acc11 = {};
    const int m    = lane & 15;
    const int ksel = (lane >> 4) << 1;

    PREFETCH_K1(0, 0);
    for (int kb = 0; kb < 8; ++kb) {
        const int cur = kb & 1;
        if (kb < 7) { PREFETCH_K1(cur ^ 1, (kb + 1) * 32); WAIT_ASYNC(6); }
        else        { WAIT_ASYNC(0); }
        __syncthreads();
#pragma unroll
        for (int k = 0; k < 32; k += 4) {
            int kk = k + ksel;
            v2f a0 = { Als[cur][wrow + m][kk],      Als[cur][wrow + m][kk + 1] };
            v2f a1 = { Als[cur][wrow + 16 + m][kk], Als[cur][wrow + 16 + m][kk + 1] };
            v2f b0 = { Bls[cur][kk][wcol + m],      Bls[cur][kk + 1][wcol + m] };
            v2f b1 = { Bls[cur][kk][wcol + 16 + m], Bls[cur][kk + 1][wcol + 16 + m] };
            acc00 = wmma_f32(a0, b0, acc00);
            acc01 = wmma_f32(a0, b1, acc01);
            acc10 = wmma_f32(a1, b0, acc10);
            acc11 = wmma_f32(a1, b1, acc11);
        }
        __syncthreads();
    }
#undef PREFETCH_K1

    const int mh = (lane >> 4) << 3;
    const int n  = lane & 15;
    const float bn0 = bias[col0 + wcol + n];
    const float bn1 = bias[col0 + wcol + 16 + n];
#pragma unroll
    for (int r = 0; r < 8; ++r) {
        int mm = r + mh;
        size_t g0 = (size_t)(row0 + wrow + mm) * ND + col0 + wcol;
        size_t g1 = (size_t)(row0 + wrow + 16 + mm) * ND + col0 + wcol;
        t1[g0 + n]      = acc00[r] + bn0;
        t1[g0 + 16 + n] = acc01[r] + bn1;
        t1[g1 + n]      = acc10[r] + bn0;
        t1[g1 + 16 + n] = acc11[r] + bn1;
    }
}

// ---------------------------------------------------------------------------
// Kernel 2: t = relu(adj[b] @ t1[b]);
//           tan_next = t * (expmap0 * proj * logmap0 scales)   [fully fused]
// block 256 thr / 8 waves: one (batch, 16-row) tile x all 256 cols.
// K-block 16, double-buffered async: 1x adj b32 + 4x t1 b128 = 5 issues/wave
// ---------------------------------------------------------------------------
__global__ void __launch_bounds__(256) aggregate_expmap(
    const float* __restrict__ adj, const float* __restrict__ t1,
    float* __restrict__ tanv)
{
    const int b    = blockIdx.y;
    const int row0 = blockIdx.x * 16;
    const int tid  = threadIdx.x;
    const int wave = tid >> 5;
    const int lane = tid & 31;
    const int wcol = wave * 32;

    __shared__ float Als[2][16][20];     // adj 16x16, stride 20
    __shared__ float Bls[2][16][272];    // t1 16x256, stride 272
    __shared__ float rowsum[16];
    __shared__ float scl[16];

    const int arr = tid >> 4, arc = tid & 15;           // adj b32 chunk
    int br[4], bc[4];
#pragma unroll
    for (int i = 0; i < 4; ++i) { int ch = tid + i * 256; br[i] = ch >> 6; bc[i] = (ch & 63) * 4; }

    const size_t adj_base = (size_t)b * NN * NN;
    const size_t t1_base  = (size_t)b * NN * ND;

#define PREFETCH_K2(buf, k0)                                                   \
    {                                                                          \
        async_b32(LDS_OFF(&Als[buf][arr][arc]),                                \
                  adj + adj_base + (size_t)(row0 + arr) * NN + (k0) + arc);    \
        _Pragma("unroll")                                                      \
        for (int i = 0; i < 4; ++i)                                            \
            async_b128(LDS_OFF(&Bls[buf][br[i]][bc[i]]),                       \
                       t1 + t1_base + (size_t)((k0) + br[i]) * ND + bc[i]);    \
    }

    v8f acc0 = {}, acc1 = {};
    const int m    = lane & 15;
    const int ksel = (lane >> 4) << 1;

    PREFETCH_K2(0, 0);
    for (int kb = 0; kb < 8; ++kb) {
        const int cur = kb & 1;
        if (kb < 7) { PREFETCH_K2(cur ^ 1, (kb + 1) * 16); WAIT_ASYNC(5); }
        else        { WAIT_ASYNC(0); }
        __syncthreads();
#pragma unroll
        for (int k = 0; k < 16; k += 4) {
            int kk = k + ksel;
            v2f a  = { Als[cur][m][kk],             Als[cur][m][kk + 1] };
            v2f b0 = { Bls[cur][kk][wcol + m],      Bls[cur][kk + 1][wcol + m] };
            v2f b1 = { Bls[cur][kk][wcol + 16 + m], Bls[cur][kk + 1][wcol + 16 + m] };
            acc0 = wmma_f32(a, b0, acc0);
            acc1 = wmma_f32(a, b1, acc1);
        }
        __syncthreads();
    }
#undef PREFETCH_K2

    // ReLU + per-row sum of squares (this wave's 32 cols)
    float ss[8];
#pragma unroll
    for (int r = 0; r < 8; ++r) {
        acc0[r] = fmaxf(acc0[r], 0.0f);
        acc1[r] = fmaxf(acc1[r], 0.0f);
        ss[r] = acc0[r] * acc0[r] + acc1[r] * acc1[r];
    }
#pragma unroll
    for (int off = 1; off < 16; off <<= 1)
#pragma unroll
        for (int r = 0; r < 8; ++r)
            ss[r] += __shfl_xor(ss[r], off, 32);

    if (tid < 16) rowsum[tid] = 0.0f;
    __syncthreads();
    if (lane == 0) {
#pragma unroll
        for (int r = 0; r < 8; ++r) atomicAdd(&rowsum[r], ss[r]);
    } else if (lane == 16) {
#pragma unroll
        for (int r = 0; r < 8; ++r) atomicAdd(&rowsum[r + 8], ss[r]);
    }
    __syncthreads();

    if (tid < 16) {
        float ntr = sqrtf(rowsum[tid]);                 // ||t||
        float nt  = fmaxf(ntr, EPSC);
        float se  = tanhf(nt) / nt;                     // expmap0 scale
        float nh  = ntr * se;                           // ||expmap0(t)||
        float sp  = (nh > MAXN) ? (MAXN / nh) : 1.0f;   // proj scale
        float nh2 = fmaxf(nh * sp, EPSC);               // ||h|| after proj
        float ls  = atanh_fast(fminf(nh2, MAXN)) / nh2; // next logmap0 scale
        scl[tid]  = se * sp * ls;                       // fully fused
    }
    __syncthreads();

    const int mh = (lane >> 4) << 3;
    const int n  = lane & 15;
#pragma unroll
    for (int r = 0; r < 8; ++r) {
        int mm = r + mh;
        float sc = scl[mm];
        size_t g = t1_base + (size_t)(row0 + mm) * ND + wcol;
        tanv[g + n]      = acc0[r] * sc;
        tanv[g + 16 + n] = acc1[r] * sc;
    }
}

// ---------------------------------------------------------------------------
// Kernel 3: out = (tan @ Wout + bout) * mask
// ---------------------------------------------------------------------------
__global__ void __launch_bounds__(256) head_gemm(
    const float* __restrict__ tanv, const float* __restrict__ Wout,
    const float* __restrict__ bout, const float* __restrict__ mask,
    float* __restrict__ out)
{
    const int tid  = threadIdx.x;
    const int wave = tid >> 5;
    const int lane = tid & 31;
    const int row0 = blockIdx.x * 128 + wave * 16;

    __shared__ float Bls[ND][20];
#pragma unroll
    for (int i = 0; i < 16; ++i) {
        int idx = tid + i * 256;
        int r = idx >> 4, c = idx & 15;
        Bls[r][c] = Wout[(size_t)r * NF + c];
    }
    __syncthreads();

    const int m    = lane & 15;
    const int ksel = (lane >> 4) << 1;
    const float* tr = tanv + (size_t)(row0 + m) * ND;

    v8f acc = {};
#pragma unroll 4
    for (int k = 0; k < ND; k += 4) {
        int kk = k + ksel;
        v2f a  = { tr[kk], tr[kk + 1] };
        v2f bb = { Bls[kk][m], Bls[kk + 1][m] };
        acc = wmma_f32(a, bb, acc);
    }

    const int mh = (lane >> 4) << 3;
    const int n  = lane & 15;
    const float bn = bout[n];
#pragma unroll
    for (int r = 0; r < 8; ++r) {
        int row = row0 + r + mh;
        float mk = mask[row];
        out[(size_t)row * NF + n] = (acc[r] + bn) * mk;
    }
}

// ---------------------------------------------------------------------------
// Host launcher
// inputs: x[B,N,D], adj[B,N,N], node_mask[B,N,1], Ws[L,D,D], bs[L,D],
//         Wout[D,F], bout[F]   -> out[B,N,F]
// ---------------------------------------------------------------------------
extern "C" void kernel_launch(void* const* d_in, const int* in_sizes, int n_in,
                              void* d_out, int out_size, void* d_ws, size_t ws_size,
                              hipStream_t stream) {
    const float* x    = (const float*)d_in[0];
    const float* adj  = (const float*)d_in[1];
    const float* mask = (const float*)d_in[2];
    const float* Ws   = (const float*)d_in[3];
    const float* bs   = (const float*)d_in[4];
    const float* Wout = (const float*)d_in[5];
    const float* bout = (const float*)d_in[6];
    float* out = (float*)d_out;

    const size_t HN = (size_t)NROWS * ND;        // 16,777,216 floats
    float* tanv = (float*)d_ws;
    float* t1   = tanv + HN;

    init_tangent<<<NROWS / 8, 256, 0, stream>>>(x, tanv);

    for (int l = 0; l < 3; ++l) {
        dim3 gA(NROWS / 64, ND / 128);           // 1024 x 2
        tangent_linear<<<gA, 256, 0, stream>>>(
            tanv, Ws + (size_t)l * ND * ND, bs + (size_t)l * ND, t1);
        dim3 gB(NN / 16, NB);                    // 8 x 512
        aggregate_expmap<<<gB, 256, 0, stream>>>(adj, t1, tanv);
    }

    head_gemm<<<NROWS / 128, 256, 0, stream>>>(tanv, Wout, bout, mask, out);
}